// MultiBoxLossFork_32710470927064
// MI455X (gfx1250) — compile-verified
//
#include <hip/hip_runtime.h>
#include <hip/hip_bf16.h>
#include <math.h>

// Problem constants (match reference)
static constexpr int N = 8, C = 20, P = 8732, M = 16;
static constexpr int NC = N * C;
static constexpr int TPB = 256;           // 8 wave32 waves per workgroup
static constexpr float THRESH = 0.5f;
static constexpr int NEG_POS_RATIO = 3;

typedef float v2f __attribute__((ext_vector_type(2)));
typedef float v8f __attribute__((ext_vector_type(8)));

// ---------------------------------------------------------------------------
// Wave-level (32 lane) sum using the CDNA5 matrix core:
// D = A(16x4 f32) * B(4x16 ones) + 0  ->  D[m][n] = rowsum_m (replicated over n)
// Lane v goes into A via VGPR0 slot; VGPR1 slot is zero, B is all ones so the
// exact (m,k)<->(lane,vgpr) striping is irrelevant: every lane's v is counted
// exactly once. d[0..7] holds 8 distinct rows per half-wave; shfl_xor(16)
// combines the two halves -> every lane holds the full 32-lane sum.
// Deterministic (fixed dataflow), requires full EXEC (callers are uniform).
// ---------------------------------------------------------------------------
__device__ __forceinline__ float wave_sum32(float v) {
    v2f a; a.x = v;    a.y = 0.0f;
    v2f b; b.x = 1.0f; b.y = 1.0f;
    v8f c = {};
    v8f d = __builtin_amdgcn_wmma_f32_16x16x4_f32(false, a, false, b,
                                                  (short)0, c, false, false);
    float s = d[0] + d[1] + d[2] + d[3] + d[4] + d[5] + d[6] + d[7];
    s += __shfl_xor(s, 16, 32);
    return s;
}

// Deterministic block (256-thread) sum: WMMA wave sums, then a fixed-order
// 8-element accumulation that every thread replays identically.
__device__ __forceinline__ float block_sum(float v, float* red8, int tid) {
    float w = wave_sum32(v);
    int wave = tid >> 5, lane = tid & 31;
    if (lane == 0) red8[wave] = w;
    __syncthreads();
    float t = 0.0f;
#pragma unroll
    for (int i = 0; i < TPB / 32; ++i) t += red8[i];
    __syncthreads();
    return t;
}

// ---------------------------------------------------------------------------
// Kernel 1: one workgroup per (n,c) row.
//  Phase 1: IoU match (argmax over objects per prior; argmax over priors per
//           object via u64 ds_max atomics), forced assignment override.
//  Phase 2: CE + L1; ce_neg row -> LDS; positives counted/summed.
//  Phase 3: top-k sum via binary search on float bit keys (exact, no sort).
// Emits per-row partials: n_pos, conf_pos, l1_sum, conf_hard.
// ---------------------------------------------------------------------------
__global__ __launch_bounds__(TPB) void multibox_row_kernel(
    const float* __restrict__ locs,    // (N,C,P,4)
    const float* __restrict__ scores,  // (N,C,P,2)
    const float* __restrict__ boxes,   // (N,C,M,4) boundary
    const int*   __restrict__ labels,  // (N,C,M)
    const float* __restrict__ priors,  // (P,4) cxcywh
    float* __restrict__ partials)      // ws: 4 arrays of NC floats
{
    __shared__ float s_ce[P];                 // ce_neg row (zeros at positives)
    __shared__ unsigned char s_obj[P];        // best object per prior
    __shared__ unsigned char s_flg[P];        // ov>=THRESH (post-override)
    __shared__ float s_bx1[M], s_by1[M], s_bx2[M], s_by2[M], s_area[M];
    __shared__ float s_cx[M], s_cy[M], s_w[M], s_h[M];
    __shared__ int s_lab[M];
    __shared__ unsigned long long s_best[M];  // packed (iou_bits<<32)|~p
    __shared__ float s_red[TPB / 32];
    __shared__ int s_cnt;
    __shared__ int s_npos;

    const int tid = threadIdx.x;
    const int nc  = blockIdx.x;               // n*C + c

    const float* __restrict__ Lb  = locs   + (size_t)nc * P * 4;
    const float* __restrict__ Sb  = scores + (size_t)nc * P * 2;
    const float* __restrict__ Bb  = boxes  + (size_t)nc * M * 4;
    const int*   __restrict__ Lab = labels + (size_t)nc * M;
    const float4* __restrict__ Pr4 = reinterpret_cast<const float4*>(priors);
    const float4* __restrict__ Lb4 = reinterpret_cast<const float4*>(Lb);
    const float2* __restrict__ Sb2 = reinterpret_cast<const float2*>(Sb);

    // ---- Phase 0: stage boxes / labels ----
    if (tid < M) {
        float x1 = Bb[tid * 4 + 0], y1 = Bb[tid * 4 + 1];
        float x2 = Bb[tid * 4 + 2], y2 = Bb[tid * 4 + 3];
        s_bx1[tid] = x1; s_by1[tid] = y1; s_bx2[tid] = x2; s_by2[tid] = y2;
        s_area[tid] = (x2 - x1) * (y2 - y1);
        s_cx[tid] = (x1 + x2) * 0.5f; s_cy[tid] = (y1 + y2) * 0.5f;
        s_w[tid]  = x2 - x1;          s_h[tid]  = y2 - y1;
        s_lab[tid]  = Lab[tid];
        s_best[tid] = 0ull;
    }
    if (tid == 0) s_npos = 0;
    __syncthreads();

    // ---- Phase 1: IoU match ----
    unsigned long long cand[M];
#pragma unroll
    for (int m = 0; m < M; ++m) cand[m] = 0ull;

    for (int p = tid; p < P; p += TPB) {
        float4 pr = Pr4[p];                               // cx,cy,w,h
        float px1 = pr.x - pr.z * 0.5f, py1 = pr.y - pr.w * 0.5f;
        float px2 = pr.x + pr.z * 0.5f, py2 = pr.y + pr.w * 0.5f;
        float areab = (px2 - px1) * (py2 - py1);          // from boundary, like ref
        float best = -1.0f; int bm = 0;
#pragma unroll
        for (int m = 0; m < M; ++m) {
            float lox = fmaxf(s_bx1[m], px1), loy = fmaxf(s_by1[m], py1);
            float hix = fminf(s_bx2[m], px2), hiy = fminf(s_by2[m], py2);
            float iw = fmaxf(hix - lox, 0.0f), ih = fmaxf(hiy - loy, 0.0f);
            float inter = iw * ih;
            float iou = inter / (s_area[m] + areab - inter);
            if (iou > best) { best = iou; bm = m; }       // first-max like argmax
            unsigned long long key =
                ((unsigned long long)__float_as_uint(iou) << 32) |
                (unsigned long long)(0xFFFFFFFFu - (unsigned)p);
            cand[m] = (key > cand[m]) ? key : cand[m];    // ties -> lowest p
        }
        s_obj[p] = (unsigned char)bm;
        s_flg[p] = (best >= THRESH) ? 1 : 0;
    }
#pragma unroll
    for (int m = 0; m < M; ++m)
        atomicMax(&s_best[m], cand[m]);                   // ds_max_u64, order-free
    __syncthreads();

    // Forced assignment: each object owns its best prior (sequential m order).
    if (tid == 0) {
        for (int m = 0; m < M; ++m) {
            unsigned p = 0xFFFFFFFFu - (unsigned)(s_best[m] & 0xFFFFFFFFull);
            s_obj[p] = (unsigned char)m;
            s_flg[p] = 1;                                 // ov := 1.0 >= THRESH
        }
    }
    __syncthreads();

    // ---- Phase 2: CE + L1 ----
    float confp = 0.0f, l1sum = 0.0f; int npl = 0;
    for (int p = tid; p < P; p += TPB) {
        int o   = s_obj[p];
        int cls = s_flg[p] ? s_lab[o] : 0;
        float2 sc = Sb2[p];
        float mx  = fmaxf(sc.x, sc.y);
        float lse = mx + logf(expf(sc.x - mx) + expf(sc.y - mx));
        float ce  = lse - ((cls != 0) ? sc.y : sc.x);     // -log_softmax[cls]
        if (cls != 0) {
            ++npl; confp += ce;
            float4 pr = Pr4[p];
            float gx = (s_cx[o] - pr.x) / (pr.z / 10.0f);
            float gy = (s_cy[o] - pr.y) / (pr.w / 10.0f);
            float gw = logf(s_w[o] / pr.z) * 5.0f;
            float gh = logf(s_h[o] / pr.w) * 5.0f;
            float4 pl = Lb4[p];
            l1sum += fabsf(pl.x - gx) + fabsf(pl.y - gy) +
                     fabsf(pl.z - gw) + fabsf(pl.w - gh);
            s_ce[p] = 0.0f;                               // ce_neg: 0 at positives
        } else {
            s_ce[p] = ce;
        }
    }
    atomicAdd(&s_npos, npl);
    float confp_t = block_sum(confp, s_red, tid);         // WMMA reduction
    float l1_t    = block_sum(l1sum, s_red, tid);
    int npos = s_npos;                                    // valid: barriers above
    __syncthreads();

    // ---- Phase 3: exact top-k sum (k = min(3*npos, P)), no sort ----
    int k = NEG_POS_RATIO * npos; if (k > P) k = P;
    float hard = 0.0f;
    if (k > 0) {                                          // uniform branch
        // ce >= 0 so float bits are order-isomorphic to values.
        unsigned lo = 0u, hi = 0x7F800000u;               // [0, +inf)
        while (hi - lo > 1u) {                            // 31 iterations
            unsigned mid = lo + ((hi - lo) >> 1);
            if (tid == 0) s_cnt = 0;
            __syncthreads();
            int cl = 0;
            for (int p = tid; p < P; p += TPB)
                cl += (__float_as_uint(s_ce[p]) >= mid) ? 1 : 0;
            atomicAdd(&s_cnt, cl);
            __syncthreads();
            int ctot = s_cnt;
            __syncthreads();
            if (ctot >= k) lo = mid; else hi = mid;
        }
        float thr = __uint_as_float(lo);                  // k-th largest value
        if (tid == 0) s_cnt = 0;
        __syncthreads();
        float sgt = 0.0f; int cgt = 0;
        for (int p = tid; p < P; p += TPB) {
            float v = s_ce[p];
            if (__float_as_uint(v) > lo) { sgt += v; ++cgt; }
        }
        atomicAdd(&s_cnt, cgt);
        float sgt_t = block_sum(sgt, s_red, tid);         // barriers inside
        int cgt_t = s_cnt;
        hard = sgt_t + (float)(k - cgt_t) * thr;          // exact w/ ties
    }

    if (tid == 0) {
        partials[0 * NC + nc] = (float)npos;
        partials[1 * NC + nc] = confp_t;
        partials[2 * NC + nc] = l1_t;
        partials[3 * NC + nc] = hard;
    }
}

// ---------------------------------------------------------------------------
// Kernel 2: combine 160 row-partials -> per-class loss -> scalar.
// Single wave32; final 20-way sum also through the WMMA reducer.
// ---------------------------------------------------------------------------
__global__ __launch_bounds__(32) void multibox_finalize_kernel(
    const float* __restrict__ partials, float* __restrict__ out)
{
    int c = threadIdx.x;                                  // 0..31, classes < C
    float np = 0.0f, cp = 0.0f, ls = 0.0f, hd = 0.0f;
    if (c < C) {
        for (int n = 0; n < N; ++n) {
            int i = n * C + c;
            np += partials[0 * NC + i];
            cp += partials[1 * NC + i];
            ls += partials[2 * NC + i];
            hd += partials[3 * NC + i];
        }
    }
    float loss_c = 0.0f;
    if (c < C && np > 0.0f) {
        float locl = ls / fmaxf(np * 4.0f, 1.0f);
        loss_c = (cp + hd + locl) / fmaxf(np, 1.0f);      // ALPHA = 1.0
    }
    loss_c *= (1.0f / (float)C);
    float tot = wave_sum32(loss_c);                       // full-EXEC wave
    if (c == 0) out[0] = tot;
}

extern "C" void kernel_launch(void* const* d_in, const int* in_sizes, int n_in,
                              void* d_out, int out_size, void* d_ws, size_t ws_size,
                              hipStream_t stream) {
    const float* locs   = (const float*)d_in[0];
    const float* scores = (const float*)d_in[1];
    const float* boxes  = (const float*)d_in[2];
    const int*   labels = (const int*)d_in[3];
    const float* priors = (const float*)d_in[4];
    float* wsf = (float*)d_ws;                            // needs 4*NC floats (2.5 KB)

    multibox_row_kernel<<<NC, TPB, 0, stream>>>(locs, scores, boxes, labels,
                                                priors, wsf);
    multibox_finalize_kernel<<<1, 32, 0, stream>>>(wsf, (float*)d_out);
}